// Seq2SeqSelfAttention_68264210202839
// MI455X (gfx1250) — compile-verified
//
#include <hip/hip_runtime.h>
#include <hip/hip_bf16.h>

typedef float v2f __attribute__((ext_vector_type(2)));
typedef float v8f __attribute__((ext_vector_type(8)));
typedef float f4  __attribute__((ext_vector_type(4)));

#define BATCH 16
#define SEQL  4096
#define NH    16
#define HD    64
#define HID   1024
#define NEGMAX -3.402823466e38f

// ---------------------------------------------------------------------------
// out[16 x 1024] = X[16 x 1024] @ W^T[1024 x 1024] + bias
// Block = 128 threads = 4 waves, split-K: wave w accumulates K in
// [w*256, w*256+256) -> 64 dependent V_WMMA_F32_16X16X4_F32 per wave
// (vs 256 serial), partials reduced through LDS.
// gridDim.x = 64 N-tiles; gridDim.y selects which of up to 3 matrices,
// so Q/K/V run concurrently in a single launch.
//
// A 16x4 f32:  lanes 0-15 -> M=lane,   v0=K0,v1=K1; lanes 16-31 -> K2,K3
// B 4x16  f32: lanes 0-15 -> N=lane,   v0=K0,v1=K1; lanes 16-31 -> K2,K3
// C/D 16x16:   lanes 0-15 -> N=lane, vi=D[M=i][N]; lanes 16-31 -> M=8+i
// ---------------------------------------------------------------------------
__global__ __launch_bounds__(128)
void gemm16_wmma_splitk(const float* __restrict__ X,
                        const float* __restrict__ W0, const float* __restrict__ B0, float* __restrict__ O0,
                        const float* __restrict__ W1, const float* __restrict__ B1, float* __restrict__ O1,
                        const float* __restrict__ W2, const float* __restrict__ B2, float* __restrict__ O2)
{
    const float* W; const float* bias; float* out;
    if      (blockIdx.y == 0) { W = W0; bias = B0; out = O0; }   // uniform branch
    else if (blockIdx.y == 1) { W = W1; bias = B1; out = O1; }
    else                      { W = W2; bias = B2; out = O2; }

    const int tid  = threadIdx.x;       // 0..127
    const int wave = tid >> 5;          // 0..3  (split-K slice)
    const int lane = tid & 31;          // EXEC all-ones within each wave
    const int half = lane >> 4;
    const int mn   = lane & 15;
    const int nbase = blockIdx.x * 16;  // 64 blocks cover N=1024

    const float* xrow = X + (size_t)mn * HID;
    const float* wrow = W + (size_t)(nbase + mn) * HID;

    const int k_begin = wave * (HID / 4);
    const int k_end   = k_begin + (HID / 4);

    v8f c = {};
    #pragma unroll 4
    for (int k0 = k_begin; k0 < k_end; k0 += 4) {
        v2f a = *(const v2f*)(xrow + k0 + 2 * half);   // 8B-aligned float2
        v2f b = *(const v2f*)(wrow + k0 + 2 * half);
        c = __builtin_amdgcn_wmma_f32_16x16x4_f32(false, a, false, b,
                                                  (short)0, c, false, false);
    }

    __shared__ float part[4][32][8];    // 4 KB split-K partials
    #pragma unroll
    for (int i = 0; i < 8; ++i) part[wave][lane][i] = c[i];
    __syncthreads();

    // 256 output elements, 128 threads -> 2 each
    for (int e = tid; e < 256; e += 128) {
        const int ln = e >> 3;
        const int i  = e & 7;
        const float s = part[0][ln][i] + part[1][ln][i]
                      + part[2][ln][i] + part[3][ln][i];
        const int hh = ln >> 4;
        const int nn = ln & 15;
        const int m  = hh * 8 + i;
        out[(size_t)m * HID + nbase + nn] = s + bias[nbase + nn];
    }
}

// ---------------------------------------------------------------------------
// Fused: KV-cache copy (with scatter of k_new/v_new at pos) + attention.
// One 256-thread block per (b,h). Bandwidth-bound: non-temporal streaming.
// ---------------------------------------------------------------------------
__global__ __launch_bounds__(256)
void attn_fused(const float* __restrict__ past_k,
                const float* __restrict__ past_v,
                const float* __restrict__ mask,
                const int*   __restrict__ cache_pos,
                const float* __restrict__ k_new,   // [16][1024] = [b][h*64+d]
                const float* __restrict__ v_new,
                const float* __restrict__ q_all,   // [16][1024]
                float* __restrict__ out_k,
                float* __restrict__ out_v,
                float* __restrict__ ctx)           // [16][1024]
{
    const int b   = blockIdx.x >> 4;
    const int h   = blockIdx.x & 15;
    const int tid = threadIdx.x;

    __shared__ float qs[HD];
    __shared__ float sc[SEQL];          // 16 KB
    __shared__ float red[256];

    const int pos = cache_pos[b];
    const size_t base = (((size_t)b * NH) + h) * SEQL * HD;
    const size_t newoff = (size_t)b * HID + h * HD;

    if (tid < HD) qs[tid] = q_all[newoff + tid];
    __syncthreads();

    // ---- Phase A: stream K slice, copy out (scatter at pos), scores -------
    float lmax = NEGMAX;
    for (int l = tid; l < SEQL; l += 256) {
        const float* krow = (l == pos) ? (k_new + newoff)
                                       : (past_k + base + (size_t)l * HD);
        float* orow = out_k + base + (size_t)l * HD;
        float s = 0.f;
        #pragma unroll
        for (int d = 0; d < HD; d += 4) {
            f4 kk = __builtin_nontemporal_load((const f4*)(krow + d));
            __builtin_nontemporal_store(kk, (f4*)(orow + d));
            s += kk.x * qs[d] + kk.y * qs[d + 1] + kk.z * qs[d + 2] + kk.w * qs[d + 3];
        }
        s *= 0.125f;                                   // 1/sqrt(64)
        if (!(mask[(size_t)b * SEQL + l] > 0.f)) s = NEGMAX;
        sc[l] = s;
        lmax = fmaxf(lmax, s);
    }

    // ---- softmax: block max ----
    red[tid] = lmax;
    __syncthreads();
    for (int off = 128; off > 0; off >>= 1) {
        if (tid < off) red[tid] = fmaxf(red[tid], red[tid + off]);
        __syncthreads();
    }
    const float gmax = red[0];
    __syncthreads();

    float lsum = 0.f;
    for (int l = tid; l < SEQL; l += 256) {
        const float e = expf(sc[l] - gmax);
        sc[l] = e;
        lsum += e;
    }
    red[tid] = lsum;
    __syncthreads();
    for (int off = 128; off > 0; off >>= 1) {
        if (tid < off) red[tid] += red[tid + off];
        __syncthreads();
    }
    const float inv = 1.0f / red[0];
    __syncthreads();

    // ---- Phase B: stream V slice, copy out, weighted accumulate ----------
    // thread -> (row-group rg in 0..3, dim d): fully coalesced (256 floats/step)
    const int d  = tid & 63;
    const int rg = tid >> 6;
    float acc = 0.f;
    for (int l = rg; l < SEQL; l += 4) {
        float vv;
        if (l == pos) vv = v_new[newoff + d];
        else          vv = __builtin_nontemporal_load(past_v + base + (size_t)l * HD + d);
        __builtin_nontemporal_store(vv, out_v + base + (size_t)l * HD + d);
        acc += sc[l] * vv;
    }
    red[tid] = acc;
    __syncthreads();
    if (tid < HD) {
        const float r = red[tid] + red[tid + 64] + red[tid + 128] + red[tid + 192];
        ctx[newoff + tid] = r * inv;   // ctx layout (b, h*D+d) matches reshape
    }
}

// ---------------------------------------------------------------------------
extern "C" void kernel_launch(void* const* d_in, const int* in_sizes, int n_in,
                              void* d_out, int out_size, void* d_ws, size_t ws_size,
                              hipStream_t stream) {
    (void)in_sizes; (void)n_in; (void)out_size; (void)ws_size;

    const float* hidden  = (const float*)d_in[0];   // (16,1,1024)
    const float* past_k  = (const float*)d_in[1];   // (16,16,4096,64)
    const float* past_v  = (const float*)d_in[2];
    const float* mask    = (const float*)d_in[3];   // (16,4096)
    const int*   cpos    = (const int*)  d_in[4];   // (16,1)
    const float* Wq      = (const float*)d_in[5];
    const float* bq      = (const float*)d_in[6];
    const float* Wk      = (const float*)d_in[7];
    const float* bk      = (const float*)d_in[8];
    const float* Wv      = (const float*)d_in[9];
    const float* bv      = (const float*)d_in[10];
    const float* Wo      = (const float*)d_in[11];
    const float* bo      = (const float*)d_in[12];

    // workspace: q, k_new, v_new, ctx  (4 x 64KB)
    float* q_ws   = (float*)d_ws;
    float* k_ws   = q_ws + BATCH * HID;
    float* v_ws   = k_ws + BATCH * HID;
    float* ctx_ws = v_ws + BATCH * HID;

    // output: attn(16K) | k_cache(67.1M) | v_cache(67.1M)
    float* out_attn = (float*)d_out;
    float* out_k    = out_attn + BATCH * HID;
    float* out_v    = out_k + (size_t)BATCH * NH * SEQL * HD;

    // Q, K, V projections fused into one launch (run concurrently)
    dim3 gqkv(64, 3);
    gemm16_wmma_splitk<<<gqkv, 128, 0, stream>>>(hidden,
                                                 Wq, bq, q_ws,
                                                 Wk, bk, k_ws,
                                                 Wv, bv, v_ws);

    attn_fused<<<BATCH * NH, 256, 0, stream>>>(past_k, past_v, mask, cpos,
                                               k_ws, v_ws, q_ws,
                                               out_k, out_v, ctx_ws);

    // Output projection
    dim3 go(64, 1);
    gemm16_wmma_splitk<<<go, 128, 0, stream>>>(ctx_ws,
                                               Wo, bo, out_attn,
                                               Wo, bo, out_attn,
                                               Wo, bo, out_attn);
}